// QuantumLayer_31112743092453
// MI455X (gfx1250) — compile-verified
//
#include <hip/hip_runtime.h>
#include <math.h>

typedef __attribute__((ext_vector_type(2))) float v2f;
typedef __attribute__((ext_vector_type(8))) float v8f;

#define NQ 12
#define NSTATE 4096

// ---------------------------------------------------------------------------
// Kernel 0: build the three 16x16 Kronecker factors of the weight RY layer.
//   A4 = R(w0)⊗..⊗R(w3), M4 = R(w4..w7), C4 = R(w8..w11)
//   R = [[c,-s],[s,c]], c=cos(w/2), s=sin(w/2). Qubit order: MSB-first.
// mats layout: float[3][16][16]  (g*256 + i*16 + j)
// ---------------------------------------------------------------------------
__global__ void build_mats_kernel(const float* __restrict__ w,
                                  float* __restrict__ mats) {
  int t = threadIdx.x;           // 0..255
  int i = t >> 4, j = t & 15;
  #pragma unroll
  for (int g = 0; g < 3; ++g) {
    float p = 1.0f;
    #pragma unroll
    for (int k = 0; k < 4; ++k) {
      int q = 4 * g + k;
      float half = 0.5f * w[q];
      float c = cosf(half), s = sinf(half);
      int ib = (i >> (3 - k)) & 1;
      int jb = (j >> (3 - k)) & 1;
      float r = ib ? (jb ? c : s) : (jb ? -s : c);
      p *= r;
    }
    mats[g * 256 + i * 16 + j] = p;
  }
}

// ---------------------------------------------------------------------------
// WMMA fragment helpers (CDNA5 ISA 7.12.2, V_WMMA_F32_16X16X4_F32, wave32).
// A 16x16x16 product = 4 chained K=4 WMMAs. frag element(k, rc):
//   p[k*sk + rc*sr]; rc = M-row (A operand) or N-col (B operand).
// ---------------------------------------------------------------------------
struct frag4 { v2f f[4]; };

__device__ __forceinline__ frag4 load_frag(const float* p, int sk, int sr,
                                           int lane) {
  frag4 r;
  const int rc  = lane & 15;
  const int off = (lane >= 16) ? 2 : 0;   // lanes 16-31 hold K+2 half
  #pragma unroll
  for (int c = 0; c < 4; ++c) {
    int k = 4 * c + off;
    r.f[c].x = p[k * sk + rc * sr];
    r.f[c].y = p[(k + 1) * sk + rc * sr];
  }
  return r;
}

__device__ __forceinline__ v8f wmma16x16x16(const frag4& A, const frag4& B) {
  v8f acc = {};
  #pragma unroll
  for (int c = 0; c < 4; ++c)
    acc = __builtin_amdgcn_wmma_f32_16x16x4_f32(
        /*neg_a=*/false, A.f[c], /*neg_b=*/false, B.f[c],
        /*c_mod=*/(short)0, acc, /*reuse_a=*/false, /*reuse_b=*/false);
  return acc;
}

__device__ __forceinline__ void store_d(float* D, int Dr, const v8f& acc,
                                        int lane) {
  const int col = lane & 15;
  const int mo  = (lane >= 16) ? 8 : 0;   // lanes 16-31 hold M=8..15 rows
  #pragma unroll
  for (int r = 0; r < 8; ++r)
    D[(r + mo) * Dr + col] = acc[r];
}

// ---------------------------------------------------------------------------
// Main kernel: one workgroup (4 wave32s) per batch element.
// ---------------------------------------------------------------------------
__global__ __launch_bounds__(128)
void qsim_kernel(const float* __restrict__ x,
                 const float* __restrict__ mats_g,
                 float* __restrict__ out) {
  __shared__ float st[NSTATE];        // state buffer A (16 KB)
  __shared__ float tp[NSTATE];        // state buffer B (16 KB)
  __shared__ float mats[3 * 256];     // A4 | M4 | C4
  __shared__ float cq[NQ], sq[NQ];
  __shared__ float prodHi[64], prodLo[64];
  __shared__ float Rrow[256];         // row sums of probs (rows = a*16+m)
  __shared__ float Pa[16], Pm[16], Pc[16];

  const int tid  = threadIdx.x;
  const int b    = blockIdx.x;
  const int lane = tid & 31;
  const int wave = tid >> 5;

  __builtin_prefetch(x + (size_t)b * NQ, 0, 1);   // global_prefetch_b8

  // Stage rotation matrices; per-sample encoding cos/sin via algebraic
  // identity: half = atan(tanh(x)) => cos = rsqrt(1+t^2), sin = t*cos.
  for (int k = tid; k < 3 * 256; k += 128) mats[k] = mats_g[k];
  if (tid < NQ) {
    float t = tanhf(x[(size_t)b * NQ + tid]);
    float r = rsqrtf(fmaf(t, t, 1.0f));
    cq[tid] = r;
    sq[tid] = t * r;
  }
  if (tid < 16) Pc[tid] = 0.0f;
  __syncthreads();

  // Half-products over qubits 0..5 (hi bits) and 6..11 (lo bits).
  if (tid < 64) {
    float p = 1.0f;
    #pragma unroll
    for (int q = 0; q < 6; ++q) p *= ((tid >> (5 - q)) & 1) ? sq[q] : cq[q];
    prodHi[tid] = p;
  } else {
    int h = tid - 64;
    float p = 1.0f;
    #pragma unroll
    for (int q = 6; q < 12; ++q) p *= ((h >> (11 - q)) & 1) ? sq[q] : cq[q];
    prodLo[h] = p;
  }
  __syncthreads();

  // Post-CNOT-chain state: amplitude at i comes from product-state index
  // j = i ^ (i>>1)  (inverse prefix-XOR, MSB-first qubit order).
  #pragma unroll
  for (int u = 0; u < NSTATE / 128; ++u) {
    int i = tid + u * 128;
    int j = i ^ (i >> 1);
    st[i] = prodHi[j >> 6] * prodLo[j & 63];
  }
  __syncthreads();

  const float* A4 = &mats[0];
  const float* M4 = &mats[256];
  const float* C4 = &mats[512];

  // Invariant rotation-matrix fragments (held in VGPRs across all tiles).
  const frag4 fC4T = load_frag(C4, 1, 16, lane);  // B[k][n] = C4[n][k]
  const frag4 fM4  = load_frag(M4, 1, 16, lane);  // A[m'][k] = M4[m'*16+k]
  const frag4 fA4  = load_frag(A4, 1, 16, lane);  // A[a'][k] = A4[a'*16+k]

  // Step 1: tp[a][m][c'] = sum_c st[a][m][c] * C4[c'][c]
  #pragma unroll
  for (int it = 0; it < 4; ++it) {
    int a = wave + 4 * it;
    frag4 fT = load_frag(&st[a * 256], 1, 16, lane);   // A[m][k]=st[..m*16+k]
    v8f acc = wmma16x16x16(fT, fC4T);
    store_d(&tp[a * 256], 16, acc, lane);
  }
  __syncthreads();

  // Step 2: st[a][m'][c'] = sum_m M4[m'][m] * tp[a][m][c']
  #pragma unroll
  for (int it = 0; it < 4; ++it) {
    int a = wave + 4 * it;
    frag4 fU = load_frag(&tp[a * 256], 16, 1, lane);   // B[k][n]=tp[..k*16+n]
    v8f acc = wmma16x16x16(fM4, fU);
    store_d(&st[a * 256], 16, acc, lane);
  }
  __syncthreads();

  // Step 3: tp[a'][t*16+n] = sum_a A4[a'][a] * st[a][t*16+n]
  #pragma unroll
  for (int it = 0; it < 4; ++it) {
    int t = wave + 4 * it;
    frag4 fV = load_frag(&st[t * 16], 256, 1, lane);   // B[k][n]=st[k*256+..n]
    v8f acc = wmma16x16x16(fA4, fV);
    store_d(&tp[t * 16], 256, acc, lane);
  }
  __syncthreads();

  // probs = psi^2. Thread tid owns rows 2*tid and 2*tid+1 (16 elems each)
  // of the 256x16 (row = a*16+m, col = c) prob matrix -> row sums w/o
  // atomics; column marginal via wave shuffle reduction + 16 adds/wave.
  {
    const int base = tid * 32;
    float r0 = 0.f, r1 = 0.f;
    float pcl[16];
    #pragma unroll
    for (int c = 0; c < 16; ++c) pcl[c] = 0.f;
    #pragma unroll
    for (int e = 0; e < 16; ++e) {
      float v0 = tp[base + e];
      float v1 = tp[base + 16 + e];
      float p0 = v0 * v0, p1 = v1 * v1;
      r0 += p0;
      r1 += p1;
      pcl[e] = p0 + p1;
    }
    Rrow[2 * tid]     = r0;
    Rrow[2 * tid + 1] = r1;
    #pragma unroll
    for (int c = 0; c < 16; ++c) {
      float v = pcl[c];
      #pragma unroll
      for (int o = 16; o > 0; o >>= 1) v += __shfl_xor(v, o, 32);
      if (lane == 0) atomicAdd(&Pc[c], v);
    }
  }
  __syncthreads();

  // Fold row sums into the a-group and m-group marginals (no atomics).
  if (tid < 16) {
    float s = 0.f;
    #pragma unroll
    for (int k = 0; k < 16; ++k) s += Rrow[tid * 16 + k];
    Pa[tid] = s;
  } else if (tid < 32) {
    int m = tid - 16;
    float s = 0.f;
    #pragma unroll
    for (int k = 0; k < 16; ++k) s += Rrow[k * 16 + m];
    Pm[m] = s;
  }
  __syncthreads();

  // <Z_q> = sum_k P_group[k] * (bit of q in k ? -1 : +1)
  if (tid < NQ) {
    int g = tid >> 2, r = tid & 3;
    const float* P = (g == 0) ? Pa : (g == 1) ? Pm : Pc;
    float o = 0.f;
    #pragma unroll
    for (int k = 0; k < 16; ++k)
      o += P[k] * (((k >> (3 - r)) & 1) ? -1.0f : 1.0f);
    out[(size_t)b * NQ + tid] = o;
  }
}

extern "C" void kernel_launch(void* const* d_in, const int* in_sizes, int n_in,
                              void* d_out, int out_size, void* d_ws, size_t ws_size,
                              hipStream_t stream) {
  const float* x = (const float*)d_in[0];   // [16384, 12] f32
  const float* w = (const float*)d_in[1];   // [36] f32 (first 12 used)
  float* out     = (float*)d_out;           // [16384, 12] f32
  float* mats    = (float*)d_ws;            // 3*256 floats

  const int B = in_sizes[0] / NQ;

  build_mats_kernel<<<1, 256, 0, stream>>>(w, mats);
  qsim_kernel<<<B, 128, 0, stream>>>(x, mats, out);
}